// MessagePassingLayer_16320875725295
// MI455X (gfx1250) — compile-verified
//
#include <hip/hip_runtime.h>
#include <hip/hip_bf16.h>

typedef __attribute__((ext_vector_type(2))) float v2f;
typedef __attribute__((ext_vector_type(8))) float v8f;

#define N_NODES 50000
#define HIDDEN 128
#define ROW_BLOCKS (N_NODES / 16)   // 3125, exact
#define WAVES_PER_BLOCK 8
#define GSTRIDE 288                 // padded group stride (floats): 256 data + 32 pad
                                    // +1 group == 32-bank rotation -> conflict-free b64

// ---------------------------------------------------------------------------
// Zero a float buffer
// ---------------------------------------------------------------------------
__global__ void zero_kernel(float* __restrict__ p, int n) {
    int i = blockIdx.x * blockDim.x + threadIdx.x;
    if (i < n) p[i] = 0.0f;
}

// ---------------------------------------------------------------------------
// Edge scatter: one wave32 per edge. Each lane loads float4 (4 features) of
// h[src] and atomically adds into agg[dst]. Lane 0 bumps degree.
// ---------------------------------------------------------------------------
__global__ void scatter_kernel(const float* __restrict__ h,
                               const int* __restrict__ src,
                               const int* __restrict__ dst,
                               float* __restrict__ agg,
                               float* __restrict__ deg,
                               int nEdges) {
    const int lane = threadIdx.x & 31;
    const int wid  = threadIdx.x >> 5;
    const int e    = blockIdx.x * (blockDim.x >> 5) + wid;
    if (e >= nEdges) return;

    const int s = src[e];
    const int d = dst[e];

    const float4* hrow = reinterpret_cast<const float4*>(h + (size_t)s * HIDDEN);
    float4 v = hrow[lane];                      // features 4*lane .. 4*lane+3

    float* arow = agg + (size_t)d * HIDDEN + lane * 4;
    atomicAdd(arow + 0, v.x);
    atomicAdd(arow + 1, v.y);
    atomicAdd(arow + 2, v.z);
    atomicAdd(arow + 3, v.w);

    if (lane == 0) atomicAdd(&deg[d], 1.0f);
}

// ---------------------------------------------------------------------------
// rdeg[i] = 1 / max(deg[i], 1)   (in place)
// ---------------------------------------------------------------------------
__global__ void rdeg_kernel(float* __restrict__ deg, int n) {
    int i = blockIdx.x * blockDim.x + threadIdx.x;
    if (i < n) deg[i] = 1.0f / fmaxf(deg[i], 1.0f);
}

// ---------------------------------------------------------------------------
// LDS fill helper: stage W [K x 128] row-major into K-paired layout
//   sW[g * GSTRIDE + col*2 + (k&1)] = W[k][col],  g = k/2
// so a B fragment (rows kb, kb+1 at column col) is one aligned ds_load_b64.
// ---------------------------------------------------------------------------
__device__ __forceinline__ void stage_W_paired(const float* __restrict__ W,
                                               float* __restrict__ sW,
                                               int nElems, int tid, int nThreads) {
    for (int idx = tid * 4; idx < nElems; idx += nThreads * 4) {
        float4 w = *reinterpret_cast<const float4*>(W + idx);
        const int k   = idx >> 7;          // / HIDDEN
        const int col = idx & (HIDDEN - 1);
        float* base = sW + (k >> 1) * GSTRIDE + (k & 1);
        base[(col + 0) * 2] = w.x;
        base[(col + 1) * 2] = w.y;
        base[(col + 2) * 2] = w.z;
        base[(col + 3) * 2] = w.w;
    }
}

// ---------------------------------------------------------------------------
// GEMM1: y1 = relu(concat(h, agg*rdeg) @ W1 + b1)     [N,256] x [256,128]
// One wave per 16-row block; computes all 8 column tiles (A reused across N).
// W1 staged in LDS, K-paired (144 KB; WGP has 320 KB). WMMA f32 16x16x4.
// ---------------------------------------------------------------------------
__global__ void __launch_bounds__(32 * WAVES_PER_BLOCK)
gemm1_kernel(const float* __restrict__ h,
             const float* __restrict__ agg,
             const float* __restrict__ rdeg,
             const float* __restrict__ W1,
             const float* __restrict__ b1,
             float* __restrict__ y1) {
    __shared__ float sW[(HIDDEN)*GSTRIDE];      // 128 K-groups * 288 = 144 KB

    const int tid = threadIdx.x;
    stage_W_paired(W1, sW, 2 * HIDDEN * HIDDEN, tid, 32 * WAVES_PER_BLOCK);
    __syncthreads();

    const int lane = tid & 31;
    const int wid  = tid >> 5;
    const int rb   = blockIdx.x * WAVES_PER_BLOCK + wid;     // 16-row block
    if (rb >= ROW_BLOCKS) return;                            // wave-uniform

    const int half = lane >> 4;          // 0: K pair {0,1}; 1: K pair {2,3}
    const int l16  = lane & 15;
    const int row  = rb * 16 + l16;      // A-matrix row for this lane
    const float rd = rdeg[row];

    v8f acc[8];
#pragma unroll
    for (int nt = 0; nt < 8; ++nt)
        acc[nt] = (v8f){0.f, 0.f, 0.f, 0.f, 0.f, 0.f, 0.f, 0.f};

    // ---- K in [0, 128): A from h -------------------------------------
    for (int k = 0; k < HIDDEN; k += 4) {
        const int kb = k + 2 * half;                 // lane's K base
        const int g  = kb >> 1;                      // paired K-group
        const float* p = h + (size_t)row * HIDDEN + kb;
        v2f a;
        a[0] = p[0];
        a[1] = p[1];
        const float* wg = sW + g * GSTRIDE;
#pragma unroll
        for (int nt = 0; nt < 8; ++nt) {
            const int col = nt * 16 + l16;
            v2f b = *reinterpret_cast<const v2f*>(wg + col * 2);
            acc[nt] = __builtin_amdgcn_wmma_f32_16x16x4_f32(
                false, a, false, b, (short)0, acc[nt], false, false);
        }
    }

    // ---- K in [128, 256): A from agg * rdeg --------------------------
    for (int k = HIDDEN; k < 2 * HIDDEN; k += 4) {
        const int kb = k + 2 * half;
        const int g  = kb >> 1;
        const float* p = agg + (size_t)row * HIDDEN + (kb - HIDDEN);
        v2f a;
        a[0] = p[0] * rd;
        a[1] = p[1] * rd;
        const float* wg = sW + g * GSTRIDE;
#pragma unroll
        for (int nt = 0; nt < 8; ++nt) {
            const int col = nt * 16 + l16;
            v2f b = *reinterpret_cast<const v2f*>(wg + col * 2);
            acc[nt] = __builtin_amdgcn_wmma_f32_16x16x4_f32(
                false, a, false, b, (short)0, acc[nt], false, false);
        }
    }

    // Epilogue: bias + relu. VGPR r of C/D holds M = r + 8*half, N = l16.
#pragma unroll
    for (int nt = 0; nt < 8; ++nt) {
        const int col = nt * 16 + l16;
        const float bias = b1[col];
#pragma unroll
        for (int r = 0; r < 8; ++r) {
            const int m = r + 8 * half;
            float v = acc[nt][r] + bias;
            y1[(size_t)(rb * 16 + m) * HIDDEN + col] = fmaxf(v, 0.0f);
        }
    }
}

// ---------------------------------------------------------------------------
// GEMM2: out = relu(y1 @ W2 + b2)      [N,128] x [128,128]
// W2 staged in LDS, K-paired (72 KB).
// ---------------------------------------------------------------------------
__global__ void __launch_bounds__(32 * WAVES_PER_BLOCK)
gemm2_kernel(const float* __restrict__ y1,
             const float* __restrict__ W2,
             const float* __restrict__ b2,
             float* __restrict__ out) {
    __shared__ float sW[(HIDDEN / 2) * GSTRIDE];    // 64 K-groups * 288 = 72 KB

    const int tid = threadIdx.x;
    stage_W_paired(W2, sW, HIDDEN * HIDDEN, tid, 32 * WAVES_PER_BLOCK);
    __syncthreads();

    const int lane = tid & 31;
    const int wid  = tid >> 5;
    const int rb   = blockIdx.x * WAVES_PER_BLOCK + wid;
    if (rb >= ROW_BLOCKS) return;                            // wave-uniform

    const int half = lane >> 4;
    const int l16  = lane & 15;
    const int row  = rb * 16 + l16;

    v8f acc[8];
#pragma unroll
    for (int nt = 0; nt < 8; ++nt)
        acc[nt] = (v8f){0.f, 0.f, 0.f, 0.f, 0.f, 0.f, 0.f, 0.f};

    for (int k = 0; k < HIDDEN; k += 4) {
        const int kb = k + 2 * half;
        const int g  = kb >> 1;
        const float* p = y1 + (size_t)row * HIDDEN + kb;
        v2f a;
        a[0] = p[0];
        a[1] = p[1];
        const float* wg = sW + g * GSTRIDE;
#pragma unroll
        for (int nt = 0; nt < 8; ++nt) {
            const int col = nt * 16 + l16;
            v2f b = *reinterpret_cast<const v2f*>(wg + col * 2);
            acc[nt] = __builtin_amdgcn_wmma_f32_16x16x4_f32(
                false, a, false, b, (short)0, acc[nt], false, false);
        }
    }

#pragma unroll
    for (int nt = 0; nt < 8; ++nt) {
        const int col = nt * 16 + l16;
        const float bias = b2[col];
#pragma unroll
        for (int r = 0; r < 8; ++r) {
            const int m = r + 8 * half;
            float v = acc[nt][r] + bias;
            out[(size_t)(rb * 16 + m) * HIDDEN + col] = fmaxf(v, 0.0f);
        }
    }
}

// ---------------------------------------------------------------------------
// Launcher
// ---------------------------------------------------------------------------
extern "C" void kernel_launch(void* const* d_in, const int* in_sizes, int n_in,
                              void* d_out, int out_size, void* d_ws, size_t ws_size,
                              hipStream_t stream) {
    const float* h    = (const float*)d_in[0];          // [50000,128]
    const int*   eidx = (const int*)d_in[1];            // [2,800000]
    const float* W1   = (const float*)d_in[2];          // [256,128]
    const float* b1   = (const float*)d_in[3];          // [128]
    const float* W2   = (const float*)d_in[4];          // [128,128]
    const float* b2   = (const float*)d_in[5];          // [128]
    float*       out  = (float*)d_out;                  // [50000,128]

    const int nEdges = in_sizes[1] / 2;
    const int* src = eidx;
    const int* dst = eidx + nEdges;

    // Workspace layout (floats): agg [N*H] | deg/rdeg [N] | y1 [N*H]
    float* agg = (float*)d_ws;
    float* deg = agg + (size_t)N_NODES * HIDDEN;
    float* y1  = deg + N_NODES;

    // 1) zero agg + deg (contiguous)
    {
        const int n = N_NODES * HIDDEN + N_NODES;
        zero_kernel<<<(n + 255) / 256, 256, 0, stream>>>(agg, n);
    }

    // 2) edge scatter-add (one wave per edge)
    {
        const int edgesPerBlock = 256 / 32;
        const int grid = (nEdges + edgesPerBlock - 1) / edgesPerBlock;
        scatter_kernel<<<grid, 256, 0, stream>>>(h, src, dst, agg, deg, nEdges);
    }

    // 3) reciprocal degree (in place)
    rdeg_kernel<<<(N_NODES + 255) / 256, 256, 0, stream>>>(deg, N_NODES);

    // 4) GEMM1 (fused concat + normalize + bias + relu), WMMA f32 16x16x4
    {
        const int grid = (ROW_BLOCKS + WAVES_PER_BLOCK - 1) / WAVES_PER_BLOCK;
        gemm1_kernel<<<grid, 32 * WAVES_PER_BLOCK, 0, stream>>>(h, agg, deg, W1, b1, y1);
    }

    // 5) GEMM2 (fused bias + relu), WMMA f32 16x16x4
    {
        const int grid = (ROW_BLOCKS + WAVES_PER_BLOCK - 1) / WAVES_PER_BLOCK;
        gemm2_kernel<<<grid, 32 * WAVES_PER_BLOCK, 0, stream>>>(y1, W2, b2, out);
    }
}